// DETRLoss_73057393705418
// MI455X (gfx1250) — compile-verified
//
#include <hip/hip_runtime.h>
#include <hip/hip_bf16.h>
#include <math.h>

// ---------------------------------------------------------------------------
// DETR loss with per-sample Hungarian (Jonker-Volgenant) matching, MI455X.
//   grid = B (one workgroup / sample), block = 256 threads (8 wave32 waves).
//   lane t owns pred column t (pred strokes in registers), GT staged in LDS.
//   JV inner-loop argmin: wave-level __shfl_xor butterfly (no barriers) +
//   8-candidate cross-wave scan => 2 workgroup barriers per path step.
//   BCE per-sample reduction on the matrix pipe: V_WMMA_F32_16X16X4_F32.
// ---------------------------------------------------------------------------

typedef __attribute__((ext_vector_type(2))) float v2f;
typedef __attribute__((ext_vector_type(8))) float v8f;

#define BN 32
#define SN 256
#define NWAVES 8
#define INF_F 3.0e38f

__global__ __launch_bounds__(256)
void detr_match_loss_kernel(const float* __restrict__ pred_strokes,   // [B,S,10]
                            const float* __restrict__ pred_validity,  // [B,S,1]
                            const float* __restrict__ targets,        // [B,S,11]
                            float* __restrict__ loss_per_b)           // [B]
{
    __shared__ float sGT[SN * 10];     // gt strokes for this sample
    __shared__ float sBCE[SN];         // per-slot BCE values
    __shared__ float sU[SN];           // row duals (rows = valid gt cols)
    __shared__ float sV[SN];           // col duals (cols = pred indices)
    __shared__ float sSP[SN];          // shortest-path distances per column
    __shared__ int   sPath[SN];        // predecessor row per column
    __shared__ int   sRow4Col[SN];
    __shared__ int   sCol4Row[SN];
    __shared__ int   sSC[SN];          // column scanned flag
    __shared__ int   sSR[SN];          // row visited flag
    __shared__ int   sCols[SN];        // compacted valid gt column indices
    __shared__ float sWVal[NWAVES];    // per-wave argmin candidates
    __shared__ int   sWIdx[NWAVES];
    __shared__ float sWSum0[NWAVES];   // per-wave partial sums
    __shared__ float sWSum1[NWAVES];
    __shared__ int   sNcols, sSink, sI;
    __shared__ float sMinVal;
    __shared__ float sAcc[4];          // {coordSum, widthSum, bcePartA, bcePartB}

    const int b    = blockIdx.x;
    const int t    = threadIdx.x;
    const int lane = t & 31;
    const int wid  = t >> 5;

    // ---- stage per-sample data -------------------------------------------
    const float* tgt = targets + (size_t)b * SN * 11;
    float gv;
    {
        const float* tr = tgt + (size_t)t * 11;
        #pragma unroll
        for (int k = 0; k < 10; ++k) sGT[t * 10 + k] = tr[k];
        gv = tr[10];
    }
    float p[10];  // this lane's pred stroke row (column t of the transposed problem)
    {
        const float* pr = pred_strokes + ((size_t)b * SN + t) * 10;
        #pragma unroll
        for (int k = 0; k < 10; ++k) p[k] = pr[k];
    }
    {   // BCE element:  -( y*max(log(p),-100) + (1-y)*max(log(1-p),-100) )
        float pv = pred_validity[(size_t)b * SN + t];
        float lp = fmaxf(logf(pv), -100.0f);
        float lq = fmaxf(logf(1.0f - pv), -100.0f);
        sBCE[t] = -(gv * lp + (1.0f - gv) * lq);
    }
    sU[t] = 0.0f; sV[t] = 0.0f;
    sRow4Col[t] = -1; sCol4Row[t] = -1;
    __syncthreads();

    // ---- compacted valid-column list (cheap serial scan) -----------------
    if (t == 0) {
        int n = 0;
        for (int j = 0; j < SN; ++j)
            if (tgt[j * 11 + 10] > 0.5f) sCols[n++] = j;
        sNcols = n;
    }
    __syncthreads();
    const int ncols = sNcols;   // nr of the transposed assignment problem (<= 256)

    // ---- Jonker-Volgenant shortest augmenting path -----------------------
    // rows r in [0,ncols): gt column sCols[r]; columns c in [0,256): pred index.
    for (int cur = 0; cur < ncols; ++cur) {
        sSP[t] = INF_F; sSC[t] = 0; sSR[t] = 0;
        if (t == 0) { sSink = -1; sI = cur; sMinVal = 0.0f; }
        __syncthreads();

        for (;;) {
            const int   i    = sI;
            const float minv = sMinVal;
            if (t == 0) sSR[i] = 1;

            const int g = sCols[i];             // gt row broadcast from LDS
            float cand = INF_F;
            if (!sSC[t]) {
                float c8 = 0.0f;
                #pragma unroll
                for (int k = 0; k < 8; ++k) c8 += fabsf(p[k] - sGT[g * 10 + k]);
                float c2 = fabsf(p[8] - sGT[g * 10 + 8]) +
                           fabsf(p[9] - sGT[g * 10 + 9]);
                float cost = 5.0f * c8 + c2;
                float r = minv + cost - sU[i] - sV[t];
                if (r < sSP[t]) { sSP[t] = r; sPath[t] = i; }
                cand = sSP[t];
            }
            // --- wave-level argmin (butterfly, EXEC-full, no barriers) ---
            float mv = cand;
            int   mi = t;
            #pragma unroll
            for (int off = 16; off > 0; off >>= 1) {
                float ov = __shfl_xor(mv, off, 32);
                int   oi = __shfl_xor(mi, off, 32);
                if (ov < mv || (ov == mv && oi < mi)) { mv = ov; mi = oi; }
            }
            if (lane == 0) { sWVal[wid] = mv; sWIdx[wid] = mi; }
            __syncthreads();
            if (t == 0) {
                float bv = sWVal[0]; int bj = sWIdx[0];
                #pragma unroll
                for (int w = 1; w < NWAVES; ++w) {
                    float wv = sWVal[w]; int wj = sWIdx[w];
                    if (wv < bv || (wv == bv && wj < bj)) { bv = wv; bj = wj; }
                }
                sMinVal = bv;
                if (sRow4Col[bj] < 0) sSink = bj;
                else { sI = sRow4Col[bj]; sSC[bj] = 1; }
            }
            __syncthreads();
            if (sSink >= 0) break;
        }

        // dual updates (pre-augmentation values, as in the reference)
        const float minv = sMinVal;
        if (t < ncols && sSR[t])
            sU[t] += (t == cur) ? minv : (minv - sSP[sCol4Row[t]]);
        if (sSC[t]) sV[t] += sSP[t] - minv;
        __syncthreads();

        // augment along predecessor chain
        if (t == 0) {
            int j = sSink;
            for (;;) {
                int i = sPath[j];
                sRow4Col[j] = i;
                int nj = sCol4Row[i];
                sCol4Row[i] = j;
                if (i == cur) break;
                j = nj;
            }
        }
        __syncthreads();
    }

    // ---- matched coord / width L1 sums (wave reduce + cross-wave) --------
    float c8 = 0.0f, c2 = 0.0f;
    if (t < ncols) {
        int pcol = sCol4Row[t];             // matched pred index for gt row t
        int g    = sCols[t];
        const float* pp = pred_strokes + ((size_t)b * SN + pcol) * 10;
        #pragma unroll
        for (int k = 0; k < 8; ++k) c8 += fabsf(pp[k] - sGT[g * 10 + k]);
        c2 = fabsf(pp[8] - sGT[g * 10 + 8]) + fabsf(pp[9] - sGT[g * 10 + 9]);
    }
    #pragma unroll
    for (int off = 16; off > 0; off >>= 1) {
        c8 += __shfl_xor(c8, off, 32);
        c2 += __shfl_xor(c2, off, 32);
    }
    if (lane == 0) { sWSum0[wid] = c8; sWSum1[wid] = c2; }
    __syncthreads();
    if (t == 0) {
        float s0 = 0.0f, s1 = 0.0f;
        #pragma unroll
        for (int w = 0; w < NWAVES; ++w) { s0 += sWSum0[w]; s1 += sWSum1[w]; }
        sAcc[0] = s0; sAcc[1] = s1;
    }

    // ---- BCE sum on the matrix pipe: 4x V_WMMA_F32_16X16X4_F32 -----------
    // A 16x4 f32 layout: lanes 0-15 hold (K=0,K=1), lanes 16-31 hold (K=2,K=3).
    // B = ones  =>  D[m,n] accumulates row sums; total = sum_m D[m,0].
    if (t < 32) {                 // whole wave 0 -> EXEC all ones inside
        const int m    = t & 15;
        const int koff = (t >> 4) << 1;
        v8f acc = {};
        v2f one; one[0] = 1.0f; one[1] = 1.0f;
        #pragma unroll
        for (int chunk = 0; chunk < 4; ++chunk) {
            v2f a;
            a[0] = sBCE[chunk * 64 + m * 4 + koff];
            a[1] = sBCE[chunk * 64 + m * 4 + koff + 1];
            acc = __builtin_amdgcn_wmma_f32_16x16x4_f32(
                false, a, false, one, (short)0, acc, false, false);
        }
        float ls = acc[0] + acc[1] + acc[2] + acc[3] +
                   acc[4] + acc[5] + acc[6] + acc[7];
        if (t == 0)  sAcc[2] = ls;   // D[0..7, 0]
        if (t == 16) sAcc[3] = ls;   // D[8..15,0]
    }
    __syncthreads();

    if (t == 0) {
        float ngt   = (float)ncols;
        float coord = sAcc[0] / fmaxf(ngt * 8.0f, 1.0f);
        float width = sAcc[1] / fmaxf(ngt * 2.0f, 1.0f);
        float bce   = (sAcc[2] + sAcc[3]) * (1.0f / (float)SN);
        loss_per_b[b] = (ncols > 0) ? (5.0f * coord + 1.0f * width + bce) : bce;
    }
}

__global__ void detr_final_mean_kernel(const float* __restrict__ lb,
                                       float* __restrict__ out)
{
    if (threadIdx.x == 0) {
        float s = 0.0f;
        for (int i = 0; i < BN; ++i) s += lb[i];
        out[0] = s * (1.0f / (float)BN);
    }
}

extern "C" void kernel_launch(void* const* d_in, const int* in_sizes, int n_in,
                              void* d_out, int out_size, void* d_ws, size_t ws_size,
                              hipStream_t stream)
{
    const float* pred_strokes  = (const float*)d_in[0];   // [32,256,10] f32
    const float* pred_validity = (const float*)d_in[1];   // [32,256,1]  f32
    const float* targets       = (const float*)d_in[2];   // [32,256,11] f32
    float* loss_per_b = (float*)d_ws;                     // 32 floats scratch

    detr_match_loss_kernel<<<BN, 256, 0, stream>>>(
        pred_strokes, pred_validity, targets, loss_per_b);
    detr_final_mean_kernel<<<1, 32, 0, stream>>>(loss_per_b, (float*)d_out);
}